// GCNNet_bench_1769526526166
// MI455X (gfx1250) — compile-verified
//
#include <hip/hip_runtime.h>
#include <hip/hip_bf16.h>
#include <math.h>

#define BN_EPS 1e-5f

typedef float v2f __attribute__((ext_vector_type(2)));
typedef float v8f __attribute__((ext_vector_type(8)));

// ---------------------------------------------------------------------------
// 1) deg = 1.0 (self loop weight)
__global__ void gcn_init_deg(float* __restrict__ deg, int n) {
    int i = blockIdx.x * blockDim.x + threadIdx.x;
    if (i < n) deg[i] = 1.0f;
}

// 2) deg[col] += ew  (scatter)
__global__ void gcn_degree(const int* __restrict__ col, const float* __restrict__ ew,
                           float* __restrict__ deg, int e) {
    int i = blockIdx.x * blockDim.x + threadIdx.x;
    if (i < e) atomicAdd(&deg[col[i]], ew[i]);
}

// 3) dinv = rsqrt(deg); re-purpose deg buffer as scalar accumulator s1 (zero it);
//    zero the 4-wide accumulator s2.
__global__ void gcn_dinv(float* __restrict__ deg_s1, float* __restrict__ dinv,
                         float* __restrict__ s2, int n) {
    int i = blockIdx.x * blockDim.x + threadIdx.x;
    if (i < n) {
        float d = rsqrtf(deg_s1[i]);   // deg >= 1 always (self loops)
        dinv[i]    = d;
        deg_s1[i]  = 0.0f;             // becomes s1 accumulator
        float4 z = make_float4(0.f, 0.f, 0.f, 0.f);
        reinterpret_cast<float4*>(s2)[i] = z;
    }
}

// 4) edge pass 1: normw[e] = dinv[r]*ew*dinv[c];  s1[c] += normw * x[r]
//    (conv1 is rank-1: only the scalar sum is needed per node)
__global__ void gcn_edge1(const int* __restrict__ row, const int* __restrict__ col,
                          const float* __restrict__ ew, const float* __restrict__ dinv,
                          const float* __restrict__ x,
                          float* __restrict__ normw, float* __restrict__ s1, int e) {
    int i = blockIdx.x * blockDim.x + threadIdx.x;
    if (i < e) {
        int r = row[i], c = col[i];
        float w = dinv[r] * ew[i] * dinv[c];
        normw[i] = w;
        atomicAdd(&s1[c], w * x[r]);
    }
}

// 5) fused node transform: t1 = s1 + dinv^2*x (self loop);
//    a1[f] = relu(BN1(W1[f]*t1 + b1[f]));  h2 = a1 @ W2   via V_WMMA_F32_16X16X4_F32.
//    One wave == one 16-node tile. A(16x4 f32): lanes0-15 -> K0,K1 ; lanes16-31 -> K2,K3.
//    B(4x16): W2 K-slice in columns 0-3, zero-padded. D(16x16): cols 0-3 = h2.
__global__ void __launch_bounds__(256)
gcn_tile_wmma(const float* __restrict__ s1, const float* __restrict__ dinv,
              const float* __restrict__ x,
              const float* __restrict__ W1, const float* __restrict__ b1,
              const float* __restrict__ g1, const float* __restrict__ be1,
              const float* __restrict__ rm1, const float* __restrict__ rv1,
              const float* __restrict__ W2,
              float* __restrict__ h2, int n, int nTiles) {
    const int lane  = threadIdx.x & 31;
    const int khalf = lane >> 4;      // 0: K{0,1}, 1: K{2,3} within each K=4 block
    const int nidx  = lane & 15;      // A-row (node) for lanes, D-column for output

    // Per-lane BN-folded affine coefficients for its 8 features, and the B matrix.
    float Af[4][2], Bf[4][2];
    v2f   Bm[4];
    #pragma unroll
    for (int kb = 0; kb < 4; ++kb) {
        #pragma unroll
        for (int j = 0; j < 2; ++j) {
            int f = 4 * kb + 2 * khalf + j;
            float inv = rsqrtf(rv1[f] + BN_EPS);
            float gs  = g1[f] * inv;
            Af[kb][j] = W1[f] * gs;
            Bf[kb][j] = (b1[f] - rm1[f]) * gs + be1[f];
            Bm[kb][j] = (nidx < 4) ? W2[f * 4 + nidx] : 0.0f;   // B[k][n], zero-padded cols 4..15
        }
    }

    const int wave   = blockIdx.x * (blockDim.x >> 5) + (threadIdx.x >> 5);
    const int nWaves = gridDim.x * (blockDim.x >> 5);

    for (int tile = wave; tile < nTiles; tile += nWaves) {   // uniform per wave: EXEC all-1 at WMMA
        int base = tile * 16;
        int node = base + nidx;
        if (node > n - 1) node = n - 1;                       // clamp (tail-safe)
        float d  = dinv[node];
        float t  = s1[node] + d * d * x[node];               // conv1 aggregate incl. self loop

        v8f acc = {};
        #pragma unroll
        for (int kb = 0; kb < 4; ++kb) {
            v2f a;
            a.x = fmaxf(Af[kb][0] * t + Bf[kb][0], 0.0f);    // a1 features (BN+ReLU fused)
            a.y = fmaxf(Af[kb][1] * t + Bf[kb][1], 0.0f);
            acc = __builtin_amdgcn_wmma_f32_16x16x4_f32(
                false, a, false, Bm[kb], (short)0, acc, false, false);
        }

        // D cols 0..3 hold h2. Uniform fast path: full tiles need no per-row guard.
        if (base + 16 <= n) {
            if (nidx < 4) {
                float* dst = h2 + (size_t)(base + 8 * khalf) * 4 + nidx;
                #pragma unroll
                for (int v = 0; v < 8; ++v) dst[v * 4] = acc[v];
            }
        } else {
            if (nidx < 4) {
                #pragma unroll
                for (int v = 0; v < 8; ++v) {
                    int m = base + v + 8 * khalf;
                    if (m < n) h2[m * 4 + nidx] = acc[v];
                }
            }
        }
    }
}

// 6) edge pass 2: s2[c][0..3] += w * h2[r][0..3]
__global__ void gcn_edge2(const int* __restrict__ row, const int* __restrict__ col,
                          const float* __restrict__ normw, const float* __restrict__ h2,
                          float* __restrict__ s2, int e) {
    int i = blockIdx.x * blockDim.x + threadIdx.x;
    if (i < e) {
        if (i + 4096 < e) {                          // stream the edge list ahead of use
            __builtin_prefetch(&row[i + 4096], 0, 0);
            __builtin_prefetch(&col[i + 4096], 0, 0);
        }
        int r = row[i], c = col[i];
        float w = normw[i];
        float4 hr = reinterpret_cast<const float4*>(h2)[r];
        atomicAdd(&s2[c * 4 + 0], w * hr.x);
        atomicAdd(&s2[c * 4 + 1], w * hr.y);
        atomicAdd(&s2[c * 4 + 2], w * hr.z);
        atomicAdd(&s2[c * 4 + 3], w * hr.w);
    }
}

// 7) node: a2 = relu(BN2(s2 + dinv^2*h2 + b2)); t3 = a2 . W3 ; zero s3 (reuses s1 buffer)
__global__ void gcn_node2(const float* __restrict__ s2, const float* __restrict__ h2,
                          const float* __restrict__ dinv,
                          const float* __restrict__ b2,
                          const float* __restrict__ g2, const float* __restrict__ be2,
                          const float* __restrict__ rm2, const float* __restrict__ rv2,
                          const float* __restrict__ W3,
                          float* __restrict__ t3, float* __restrict__ s3, int n) {
    int i = blockIdx.x * blockDim.x + threadIdx.x;
    if (i < n) {
        float d  = dinv[i];
        float dd = d * d;
        float4 sv = reinterpret_cast<const float4*>(s2)[i];
        float4 hv = reinterpret_cast<const float4*>(h2)[i];
        float acc = 0.0f;
        float sj[4] = {sv.x, sv.y, sv.z, sv.w};
        float hj[4] = {hv.x, hv.y, hv.z, hv.w};
        #pragma unroll
        for (int j = 0; j < 4; ++j) {
            float v = sj[j] + dd * hj[j] + b2[j];
            float a = (v - rm2[j]) * rsqrtf(rv2[j] + BN_EPS) * g2[j] + be2[j];
            a = fmaxf(a, 0.0f);
            acc += a * W3[j];                                // conv3 rank-1: project now
        }
        t3[i] = acc;
        s3[i] = 0.0f;                                        // accumulator for edge pass 3
    }
}

// 8) edge pass 3: s3[c] += w * t3[r]
__global__ void gcn_edge3(const int* __restrict__ row, const int* __restrict__ col,
                          const float* __restrict__ normw, const float* __restrict__ t3,
                          float* __restrict__ s3, int e) {
    int i = blockIdx.x * blockDim.x + threadIdx.x;
    if (i < e) {
        atomicAdd(&s3[col[i]], normw[i] * t3[row[i]]);
    }
}

// 9) final: BN3, Linear(1,1), sigmoid
__global__ void gcn_final(const float* __restrict__ s3, const float* __restrict__ t3,
                          const float* __restrict__ dinv,
                          const float* __restrict__ b3,
                          const float* __restrict__ g3, const float* __restrict__ be3,
                          const float* __restrict__ rm3, const float* __restrict__ rv3,
                          const float* __restrict__ Wl, const float* __restrict__ bl,
                          float* __restrict__ out, int n) {
    int i = blockIdx.x * blockDim.x + threadIdx.x;
    if (i < n) {
        float d  = dinv[i];
        float v  = s3[i] + d * d * t3[i] + b3[0];
        float h  = (v - rm3[0]) * rsqrtf(rv3[0] + BN_EPS) * g3[0] + be3[0];
        float y  = h * Wl[0] + bl[0];
        out[i] = 1.0f / (1.0f + expf(-y));
    }
}

// ---------------------------------------------------------------------------
extern "C" void kernel_launch(void* const* d_in, const int* in_sizes, int n_in,
                              void* d_out, int out_size, void* d_ws, size_t ws_size,
                              hipStream_t stream) {
    const float* x   = (const float*)d_in[0];
    const int*   ei  = (const int*)  d_in[1];
    const float* ew  = (const float*)d_in[2];
    const float* W1  = (const float*)d_in[3];
    const float* b1  = (const float*)d_in[4];
    const float* W2  = (const float*)d_in[5];
    const float* b2  = (const float*)d_in[6];
    const float* W3  = (const float*)d_in[7];
    const float* b3  = (const float*)d_in[8];
    const float* g1  = (const float*)d_in[9];
    const float* be1 = (const float*)d_in[10];
    const float* rm1 = (const float*)d_in[11];
    const float* rv1 = (const float*)d_in[12];
    const float* g2  = (const float*)d_in[13];
    const float* be2 = (const float*)d_in[14];
    const float* rm2 = (const float*)d_in[15];
    const float* rv2 = (const float*)d_in[16];
    const float* g3  = (const float*)d_in[17];
    const float* be3 = (const float*)d_in[18];
    const float* rm3 = (const float*)d_in[19];
    const float* rv3 = (const float*)d_in[20];
    const float* Wl  = (const float*)d_in[21];
    const float* bl  = (const float*)d_in[22];

    const int N = in_sizes[0];          // 100000
    const int E = in_sizes[1] / 2;      // 3200000
    const int* row = ei;
    const int* col = ei + E;

    // Workspace layout (floats). Total = 11*N + E floats (~17.2 MB).
    float* wsf   = (float*)d_ws;
    float* dinv  = wsf;                 // [N]
    float* degS  = wsf + (size_t)N;     // [N]  deg -> s1 -> s3 (sequentially reused)
    float* h2    = wsf + (size_t)2 * N; // [4N] (16B aligned per node)
    float* s2    = wsf + (size_t)6 * N; // [4N]
    float* t3    = wsf + (size_t)10 * N;// [N]
    float* normw = wsf + (size_t)11 * N;// [E]

    const int TB = 256;
    const int nodeBlocks = (N + TB - 1) / TB;
    const int edgeBlocks = (E + TB - 1) / TB;
    const int nTiles     = (N + 15) / 16;
    const int wavesPerBlock = TB / 32;
    const int tileBlocks = (nTiles + wavesPerBlock - 1) / wavesPerBlock;

    gcn_init_deg<<<nodeBlocks, TB, 0, stream>>>(degS, N);
    gcn_degree  <<<edgeBlocks, TB, 0, stream>>>(col, ew, degS, E);
    gcn_dinv    <<<nodeBlocks, TB, 0, stream>>>(degS, dinv, s2, N);
    gcn_edge1   <<<edgeBlocks, TB, 0, stream>>>(row, col, ew, dinv, x, normw, degS, E);
    gcn_tile_wmma<<<tileBlocks, TB, 0, stream>>>(degS, dinv, x,
                                                 W1, b1, g1, be1, rm1, rv1, W2,
                                                 h2, N, nTiles);
    gcn_edge2   <<<edgeBlocks, TB, 0, stream>>>(row, col, normw, h2, s2, E);
    gcn_node2   <<<nodeBlocks, TB, 0, stream>>>(s2, h2, dinv, b2, g2, be2, rm2, rv2,
                                                W3, t3, degS, N);
    gcn_edge3   <<<edgeBlocks, TB, 0, stream>>>(row, col, normw, t3, degS, E);
    gcn_final   <<<nodeBlocks, TB, 0, stream>>>(degS, t3, dinv, b3, g3, be3, rm3, rv3,
                                                Wl, bl, (float*)d_out, N);
}